// EwaldElectrostaticEnergy_26534307955289
// MI455X (gfx1250) — compile-verified
//
#include <hip/hip_runtime.h>
#include <hip/hip_bf16.h>
#include <math.h>

// ---------------------------------------------------------------------------
// Ewald electrostatics for MI455X (gfx1250).
// Reciprocal part re-factored as a complex GEMM (M=16 ku, N=448 kv/kw pairs,
// K=1008 atoms per graph) executed with V_WMMA_F32_16X16X4_F32.
// A-operands (q*e^{i2pi ku f0}) stored pair-interleaved so each A-fragment is
// a single aligned b64 load straight into the WMMA source registers.
// B tables (cv, cw) are interleaved float2 records. ~26 MB, L2-resident.
// ---------------------------------------------------------------------------

#define NG      50      // graphs
#define APG     1000    // atoms per graph
#define SLOTS   1008    // atoms padded to multiple of 16 (K padding, zeros)
#define NPAIRS  (SLOTS/2)
#define BREC    48      // B record: [0,24) cv (21+pad), [24,48) cw (21+pad)
#define NTILES  28      // ceil(441/16) column tiles over (kv,kw) pairs
#define KE_C    14.399645351950548f
#define TWOPI   6.28318530717958647692f
#define INV_SQRT_PI 0.5641895835477563f

typedef __attribute__((ext_vector_type(2))) float v2f;
typedef __attribute__((ext_vector_type(8))) float v8f;

// workspace layout (float offsets)
#define WS_EREAL  0                         // 50000 floats (+pad)
#define WS_ENERGY 50048                     // 64 floats
#define WS_CELL   50112                     // 50 * 12: inv(cell)[9], |det|, pad
#define WS_ARE    50752                     // NG*NPAIRS*16*2 floats (806,400)
#define ARE_FLTS  (NG * NPAIRS * 16 * 2)
#define WS_AIM    (WS_ARE + ARE_FLTS)
#define WS_TABB   (WS_AIM + ARE_FLTS)       // float2[NG*SLOTS*BREC]

// ---------------------------------------------------------------------------
// Kernel 0: per-graph 3x3 cell inverse + volume.
// ---------------------------------------------------------------------------
__global__ void k_cellinv(const float* __restrict__ cell, float* __restrict__ ws) {
    int g = blockIdx.x * blockDim.x + threadIdx.x;
    if (g >= NG) return;
    const float* c = cell + g * 9;
    float c00=c[0],c01=c[1],c02=c[2],c10=c[3],c11=c[4],c12=c[5],c20=c[6],c21=c[7],c22=c[8];
    float det = c00*(c11*c22 - c12*c21) - c01*(c10*c22 - c12*c20) + c02*(c10*c21 - c11*c20);
    float id  = 1.0f / det;
    float* o = ws + WS_CELL + g * 12;
    o[0]=(c11*c22-c12*c21)*id; o[1]=(c02*c21-c01*c22)*id; o[2]=(c01*c12-c02*c11)*id;
    o[3]=(c12*c20-c10*c22)*id; o[4]=(c00*c22-c02*c20)*id; o[5]=(c02*c10-c00*c12)*id;
    o[6]=(c10*c21-c11*c20)*id; o[7]=(c01*c20-c00*c21)*id; o[8]=(c00*c11-c01*c10)*id;
    o[9]=fabsf(det);
}

// ---------------------------------------------------------------------------
// Kernel 1: per-atom complex phase tables. One 64-thread block per atom slot,
// one sincos per thread (2.7M total). e<16 -> A planes (pair-interleaved),
// e>=16 -> B record (interleaved float2).
// ---------------------------------------------------------------------------
__global__ void k_tables(const float* __restrict__ pos, const float* __restrict__ q,
                         const float* __restrict__ ws, float* __restrict__ aRe,
                         float* __restrict__ aIm, float2* __restrict__ tabB) {
    int slot = blockIdx.x;           // 0 .. NG*SLOTS-1
    int e    = threadIdx.x;          // 0 .. 63
    int g    = slot / SLOTS;
    int s    = slot - g * SLOTS;
    bool valid = (s < APG);
    float f0 = 0.f, f1 = 0.f, f2 = 0.f, qa = 0.f;
    if (valid) {
        int a = g * APG + s;
        const float* inv = ws + WS_CELL + g * 12;
        float px = pos[a*3+0], py = pos[a*3+1], pz = pos[a*3+2];
        // fractional coords: f_i = sum_j pos_j * inv[j][i]
        f0 = px*inv[0] + py*inv[3] + pz*inv[6];
        f1 = px*inv[1] + py*inv[4] + pz*inv[7];
        f2 = px*inv[2] + py*inv[5] + pz*inv[8];
        qa = q[a];
    }
    if (e < 16) {
        float re = 0.f, im = 0.f;
        if (valid && e <= 10) {
            float sn, cs;
            __sincosf(TWOPI * (float)e * f0, &sn, &cs);
            re = qa * cs; im = qa * sn;
        }
        long idx = (((long)g * NPAIRS + (s >> 1)) * 16 + e) * 2 + (s & 1);
        aRe[idx] = re;
        aIm[idx] = im;
    } else {
        int ep = e - 16;             // 0..47
        float re = 0.f, im = 0.f;
        if (valid) {
            float f; int nn; bool inrange;
            if (ep < 24) { f = f1; nn = ep - 10; inrange = (ep <= 20); }
            else         { f = f2; nn = ep - 34; inrange = (ep <= 44); }
            if (inrange) {
                float sn, cs;
                __sincosf(TWOPI * (float)nn * f, &sn, &cs);
                re = cs; im = sn;
            }
        }
        tabB[(long)slot * BREC + ep] = make_float2(re, im);
    }
}

// ---------------------------------------------------------------------------
// Kernel 2: real-space pairwise term, scatter-add into e_real.
// ---------------------------------------------------------------------------
__global__ void k_edges(const int* __restrict__ ei, const int* __restrict__ ej,
                        const float* __restrict__ len, const float* __restrict__ q,
                        float* __restrict__ e_real, int ne) {
    int i = blockIdx.x * blockDim.x + threadIdx.x;
    if (i >= ne) return;
    float r = len[i];
    if (r > 5.0f) return;
    int ii = ei[i], jj = ej[i];
    float fac = 0.5f * KE_C * q[ii] * q[jj];
    float pc = 0.f;
    if (r < 4.0f) {  // poly cutoff, p=6: 1 - 28u^6 + 48u^7 - 21u^8
        float u = r * 0.25f;
        float u2 = u*u, u4 = u2*u2, u6 = u4*u2;
        pc = 1.f - 28.f*u6 + 48.f*u6*u - 21.f*u6*u2;
    }
    float pe = fac / r * (erfcf(r) - pc);
    atomicAdd(&e_real[ii], pe);
}

// ---------------------------------------------------------------------------
// Kernel 3: structure factor as complex GEMM via V_WMMA_F32_16X16X4_F32.
// One wave32 per (graph, 16-column tile). M rows = ku 0..15 (11 valid),
// N cols = (kv,kw) pair index, K = 1008 atom slots in steps of 4.
// Per K-step: 2 aligned b64 A-loads (direct v2f) + 4 b64 B-loads + 2 complex
// products + 4 WMMAs. Complex result: S_r = RR - II, S_i = RI + IR.
// Epilogue folds k_factor * sym * |S|^2 and reduces to energy[g].
// ---------------------------------------------------------------------------
__global__ void __launch_bounds__(32)
k_sfac(const float* __restrict__ aRe, const float* __restrict__ aIm,
       const float2* __restrict__ tabB, const float* __restrict__ ws,
       float* __restrict__ energy) {
    int g    = blockIdx.y;
    int tile = blockIdx.x;
    int lane = threadIdx.x;
    int col  = lane & 15;
    int half = lane >> 4;
    int n    = tile * 16 + col;          // pair index, up to 447 (>=441 is padding)
    int mi   = n / 21, pi = n - mi * 21; // kv,kw table indices (mi==21 hits zero pad)
    long bA  = (long)g * NPAIRS * 32;    // float index into A planes
    long bB  = (long)g * SLOTS * BREC;   // float2 index into B table

    v8f accRR = {0,0,0,0,0,0,0,0};
    v8f accII = {0,0,0,0,0,0,0,0};
    v8f accRI = {0,0,0,0,0,0,0,0};
    v8f accIR = {0,0,0,0,0,0,0,0};

    for (int a0 = 0; a0 < SLOTS; a0 += 4) {
        // A fragment: A[M=col][K=2*half+j]  B fragment: B[K=2*half+j][N=col]
        long aoff = bA + ((long)((a0 >> 1) + half)) * 32 + col * 2;
        v2f Ar = *(const v2f*)(aRe + aoff);
        v2f Ai = *(const v2f*)(aIm + aoff);
        long r0 = bB + (long)(a0 + 2 * half) * BREC;
        float2 v0 = tabB[r0 + mi];            // e^{i 2pi kv f1}, atom j=0
        float2 w0 = tabB[r0 + 24 + pi];       // e^{i 2pi kw f2}, atom j=0
        float2 v1 = tabB[r0 + BREC + mi];     // atom j=1
        float2 w1 = tabB[r0 + BREC + 24 + pi];
        // b = cv * cw (complex)
        float br0 = v0.x*w0.x - v0.y*w0.y, bi0 = v0.x*w0.y + v0.y*w0.x;
        float br1 = v1.x*w1.x - v1.y*w1.y, bi1 = v1.x*w1.y + v1.y*w1.x;
        v2f Br = {br0, br1}, Bi = {bi0, bi1};
        accRR = __builtin_amdgcn_wmma_f32_16x16x4_f32(false, Ar, false, Br, (short)0, accRR, false, false);
        accII = __builtin_amdgcn_wmma_f32_16x16x4_f32(false, Ai, false, Bi, (short)0, accII, false, false);
        accRI = __builtin_amdgcn_wmma_f32_16x16x4_f32(false, Ar, false, Bi, (short)0, accRI, false, false);
        accIR = __builtin_amdgcn_wmma_f32_16x16x4_f32(false, Ai, false, Br, (short)0, accIR, false, false);
    }

    const float* inv = ws + WS_CELL + g * 12;
    float i0=inv[0],i1=inv[1],i2=inv[2],i3=inv[3],i4=inv[4],i5=inv[5],i6=inv[6],i7=inv[7],i8=inv[8];
    float fm = (float)(mi - 10), fp = (float)(pi - 10);
    int   mp2 = (mi - 10)*(mi - 10) + (pi - 10)*(pi - 10);
    float lsum = 0.f;
    #pragma unroll
    for (int r = 0; r < 8; ++r) {
        int ku = r + 8 * half;               // C/D layout: VGPR r, lane half -> M row
        float sr = accRR[r] - accII[r];
        float si = accRI[r] + accIR[r];
        int ksqi = ku*ku + mp2;
        if (n < 441 && ku <= 10 && ksqi > 0 && ksqi <= 100) {
            float fku = (float)ku;
            float kx = TWOPI * (fku*i0 + fm*i1 + fp*i2);
            float ky = TWOPI * (fku*i3 + fm*i4 + fp*i5);
            float kz = TWOPI * (fku*i6 + fm*i7 + fp*i8);
            float ksq = kx*kx + ky*ky + kz*kz;
            float kfac = __expf(-0.25f * ksq) / ksq;
            float sym  = (ku == 0) ? 1.f : 2.f;
            lsum += kfac * sym * (sr*sr + si*si);
        }
    }
    #pragma unroll
    for (int o = 16; o > 0; o >>= 1) lsum += __shfl_xor(lsum, o, 32);
    if (lane == 0) atomicAdd(&energy[g], lsum);
}

// ---------------------------------------------------------------------------
// Kernel 4: combine per-atom output.
// ---------------------------------------------------------------------------
__global__ void k_final(const float* __restrict__ ae, const float* __restrict__ q,
                        const int* __restrict__ batch, const int* __restrict__ n_atoms,
                        const float* __restrict__ ws, float* __restrict__ out, int na) {
    int a = blockIdx.x * blockDim.x + threadIdx.x;
    if (a >= na) return;
    int g = batch[a];
    float vol = ws[WS_CELL + g * 12 + 9];
    float eg  = ws[WS_ENERGY + g] * (TWOPI * KE_C) / vol / (float)n_atoms[g];
    float qa  = q[a];
    out[a] = ae[a] + ws[WS_EREAL + a] + eg - KE_C * INV_SQRT_PI * qa * qa;
}

// ---------------------------------------------------------------------------
extern "C" void kernel_launch(void* const* d_in, const int* in_sizes, int n_in,
                              void* d_out, int out_size, void* d_ws, size_t ws_size,
                              hipStream_t stream) {
    const int*   edge      = (const int*)d_in[0];
    int          ne        = in_sizes[0] / 2;
    const float* lengths   = (const float*)d_in[1];
    const float* positions = (const float*)d_in[2];
    const float* cell      = (const float*)d_in[3];
    const int*   batch     = (const int*)d_in[4];
    const int*   n_atoms   = (const int*)d_in[5];
    const float* q         = (const float*)d_in[6];
    const float* ae        = (const float*)d_in[7];

    float*  ws     = (float*)d_ws;
    float*  e_real = ws + WS_EREAL;
    float*  energy = ws + WS_ENERGY;
    float*  aRe    = ws + WS_ARE;
    float*  aIm    = ws + WS_AIM;
    float2* tabB   = (float2*)(ws + WS_TABB);

    // zero e_real + energy accumulators
    hipMemsetAsync(ws, 0, (size_t)WS_CELL * sizeof(float), stream);

    k_cellinv<<<1, 64, 0, stream>>>(cell, ws);
    k_tables<<<NG * SLOTS, 64, 0, stream>>>(positions, q, ws, aRe, aIm, tabB);
    k_edges<<<(ne + 255) / 256, 256, 0, stream>>>(edge, edge + ne, lengths, q, e_real, ne);
    dim3 gg(NTILES, NG);
    k_sfac<<<gg, 32, 0, stream>>>(aRe, aIm, tabB, ws, energy);
    k_final<<<(out_size + 255) / 256, 256, 0, stream>>>(ae, q, batch, n_atoms, ws,
                                                        (float*)d_out, out_size);
}